// SpatialConsistencyLoss_84370337562972
// MI455X (gfx1250) — compile-verified
//
#include <hip/hip_runtime.h>
#include <hip/hip_bf16.h>
#include <stdint.h>

// Stage 1: pooled-gray difference.
// d[b,i,j] = mean_{c,4x4}(x) - mean_{c,4x4}(ex), shape (16,128,128).
// One block per (b, pooled_row). 24 staged rows (2 inputs * 3 ch * 4 h-rows)
// of 512 floats each = 48 KB LDS, copied with async global->LDS B128 ops.
__global__ __launch_bounds__(128) void pool_diff_kernel(
    const float* __restrict__ x, const float* __restrict__ ex,
    float* __restrict__ d) {
  __shared__ __align__(16) float sh[24 * 512];

  const int t  = threadIdx.x;          // 0..127 : pooled column
  const int b  = blockIdx.x >> 7;      // batch
  const int pr = blockIdx.x & 127;     // pooled row

  // LDS byte offset of sh within this workgroup's allocation
  // (flat shared pointers have the LDS offset in their low 32 bits).
  const unsigned shBase = (unsigned)(uintptr_t)(&sh[0]);

  // Issue 24 row copies: each thread moves 16B of every 2KB row.
  // Row index r24: [0,12) from x, [12,24) from ex; within input: c*4 + r.
  #pragma unroll
  for (int r24 = 0; r24 < 24; ++r24) {
    const int s  = r24 / 12;           // which input
    const int cr = r24 % 12;
    const int c  = cr >> 2;            // channel 0..2
    const int r  = cr & 3;             // sub-row 0..3
    const float* src = (s == 0) ? x : ex;
    const float* g = src
        + (((size_t)b * 3 + (size_t)c) * 512 + (size_t)(4 * pr + r)) * 512
        + (size_t)(t * 4);
    const unsigned loff = shBase + (unsigned)(r24 * 512 + t * 4) * 4u;
    asm volatile("global_load_async_to_lds_b128 %0, %1, off"
                 :: "v"(loff), "v"(g) : "memory");
  }

  // Each wave waits for its own async copies, then workgroup barrier.
  asm volatile("s_wait_asynccnt 0" ::: "memory");
  __syncthreads();

  // Thread t reduces pooled column t: 12 rows * float4 per input.
  float sx = 0.0f, se = 0.0f;
  #pragma unroll
  for (int j = 0; j < 12; ++j) {
    float4 v = *(const float4*)&sh[j * 512 + 4 * t];
    sx += (v.x + v.y) + (v.z + v.w);
  }
  #pragma unroll
  for (int j = 12; j < 24; ++j) {
    float4 v = *(const float4*)&sh[j * 512 + 4 * t];
    se += (v.x + v.y) + (v.z + v.w);
  }

  d[(size_t)blockIdx.x * 128 + t] = (sx - se) * (1.0f / 48.0f);
}

// Stage 2: 5-point stencil loss on the 1MB difference map (L2-resident).
// out = (d-dL)^2 + (d-dR)^2 + (d-dU)^2 + (d-dD)^2, zero-padded neighbors.
__global__ __launch_bounds__(256) void dir_loss_kernel(
    const float* __restrict__ d, float* __restrict__ out) {
  const int idx = blockIdx.x * 256 + threadIdx.x;   // 0..262143
  const int j = idx & 127;
  const int i = (idx >> 7) & 127;
  const int b = idx >> 14;
  const float* pd = d + ((size_t)b << 14);

  const float c  = pd[(i << 7) | j];
  const float l  = (j > 0)   ? pd[(i << 7) | (j - 1)] : 0.0f;
  const float r  = (j < 127) ? pd[(i << 7) | (j + 1)] : 0.0f;
  const float u  = (i > 0)   ? pd[((i - 1) << 7) | j] : 0.0f;
  const float dn = (i < 127) ? pd[((i + 1) << 7) | j] : 0.0f;

  const float a0 = c - l, a1 = c - r, a2 = c - u, a3 = c - dn;
  out[idx] = a0 * a0 + a1 * a1 + a2 * a2 + a3 * a3;
}

extern "C" void kernel_launch(void* const* d_in, const int* in_sizes, int n_in,
                              void* d_out, int out_size, void* d_ws, size_t ws_size,
                              hipStream_t stream) {
  const float* x  = (const float*)d_in[0];
  const float* ex = (const float*)d_in[1];
  float* out  = (float*)d_out;
  float* dbuf = (float*)d_ws;     // 16*128*128 floats = 1 MB scratch

  // 16 batches * 128 pooled rows, one block each.
  pool_diff_kernel<<<dim3(16 * 128), dim3(128), 0, stream>>>(x, ex, dbuf);
  // 262144 outputs.
  dir_loss_kernel<<<dim3(1024), dim3(256), 0, stream>>>(dbuf, out);
}